// EWMAPredictor_46548855554489
// MI455X (gfx1250) — compile-verified
//
#include <hip/hip_runtime.h>

typedef __attribute__((ext_vector_type(2))) float v2f;
typedef __attribute__((ext_vector_type(8))) float v8f;

#define VOLA_WIN   128
#define LOOK_BACK  524288
#define CHUNK      1024          // windows per block
#define TPB        256           // 8 waves of 32
#define SPAN       (CHUNK + VOLA_WIN)   // 1152 staged elements per block

// Kernel 1: each block computes the weighted-variance partial sum for
// CHUNK consecutive EWMA indices. Window i covers x[n-1-W-i .. n-2-i].
__global__ __launch_bounds__(TPB) void ewma_var_partial(
    const float* __restrict__ x,
    const float* __restrict__ raw_ff,
    float* __restrict__ partial,
    int n)
{
    __shared__ float lds[SPAN];
    __shared__ float wavesum[TPB / 32];

    const int b = blockIdx.x;
    const int t = threadIdx.x;

    // forgetting factor (scalar broadcast; cheap)
    const float ff   = 1.0f / (1.0f + expf(-raw_ff[0]));
    const float l2ff = log2f(ff);

    // Block b owns i in [b*CHUNK, b*CHUNK+CHUNK).
    // Union of its windows = x[base .. base+SPAN-2]; stage SPAN elements.
    const int base = n - SPAN - b * CHUNK;   // == 0 for the last block

    __builtin_prefetch(&x[base], 0, 0);      // global_prefetch_b8 (gfx1250)
    for (int k = t; k < SPAN; k += TPB)
        lds[k] = x[base + k];
    __syncthreads();

    float acc = 0.0f;
#pragma unroll
    for (int r = 0; r < CHUNK / TPB; ++r) {
        const int ii  = t + r * TPB;             // local window id
        const int off = (CHUNK - 1) - ii;        // window start within lds
        float s = 0.0f, s2 = 0.0f;
#pragma unroll 8
        for (int j = 0; j < VOLA_WIN; ++j) {
            const float v = lds[off + j];
            s  += v;
            s2  = fmaf(v, v, s2);
        }
        // unbiased variance: (sum(x^2) - sum(x)^2/W) / (W-1)
        const float var = (s2 - s * s * (1.0f / VOLA_WIN)) *
                          (1.0f / (VOLA_WIN - 1));
        const int   i   = b * CHUNK + ii;
        const float w   = exp2f((float)i * l2ff);  // underflows to 0 like ref
        acc = fmaf(w, var, acc);
    }

    // ---- wave-level reduction via V_WMMA_F32_16X16X4_F32 ----
    // A (16x4 f32, 2 VGPRs): VGPR0 holds K=0 (lanes 0-15) / K=2 (lanes 16-31),
    // VGPR1 holds K=1 / K=3. Put each lane's value in VGPR0, zero VGPR1.
    // With B = all-ones, D[m][n] = A[m][0] + A[m][2] = v_m + v_{m+16}.
    v2f a;  a.x = acc;  a.y = 0.0f;
    v2f bm; bm.x = 1.0f; bm.y = 1.0f;
    v8f c = {};
    v8f d = __builtin_amdgcn_wmma_f32_16x16x4_f32(
        /*neg_a=*/false, a, /*neg_b=*/false, bm,
        /*c_mod=*/(short)0, c, /*reuse_a=*/false, /*reuse_b=*/false);

    // Per lane: D VGPRs 0..7 hold rows 0..7 (lanes 0-15) or rows 8..15
    // (lanes 16-31). Sum them, then fold the two half-sums with one xor-shuffle.
    float h = d[0] + d[1] + d[2] + d[3] + d[4] + d[5] + d[6] + d[7];
    h += __shfl_xor(h, 16, 32);                 // full wave(32) sum in all lanes

    const int lane = t & 31, wave = t >> 5;
    if (lane == 0) wavesum[wave] = h;
    __syncthreads();
    if (t == 0) {
        float tot = 0.0f;
#pragma unroll
        for (int w8 = 0; w8 < TPB / 32; ++w8) tot += wavesum[w8];
        partial[b] = tot;
    }
}

// Kernel 2: reduce the per-block partials and apply the normalizer.
__global__ __launch_bounds__(256) void ewma_finalize(
    const float* __restrict__ partial,
    const float* __restrict__ raw_ff,
    float* __restrict__ out,
    int nb)
{
    __shared__ float red[256];
    const int t = threadIdx.x;
    float s = 0.0f;
    for (int k = t; k < nb; k += 256) s += partial[k];
    red[t] = s;
    __syncthreads();
    for (int stride = 128; stride > 0; stride >>= 1) {
        if (t < stride) red[t] += red[t + stride];
        __syncthreads();
    }
    if (t == 0) {
        const float ff   = 1.0f / (1.0f + expf(-raw_ff[0]));
        const float ffL  = exp2f((float)LOOK_BACK * log2f(ff)); // -> 0 typically
        out[0] = red[0] * (1.0f - ff) / (1.0f - ffL);
    }
}

extern "C" void kernel_launch(void* const* d_in, const int* in_sizes, int n_in,
                              void* d_out, int out_size, void* d_ws, size_t ws_size,
                              hipStream_t stream) {
    const float* x   = (const float*)d_in[0];   // past_returns [N]
    // d_in[1] = features (unused by the module)
    const float* raw = (const float*)d_in[2];   // raw_forgetting_factor [1]
    float* out       = (float*)d_out;
    float* partials  = (float*)d_ws;            // 512 floats of scratch

    const int n        = in_sizes[0];           // 524416
    const int nblocks  = LOOK_BACK / CHUNK;     // 512

    ewma_var_partial<<<nblocks, TPB, 0, stream>>>(x, raw, partials, n);
    ewma_finalize<<<1, 256, 0, stream>>>(partials, raw, out, nblocks);
}